// WeightedBiasEncoder_59596966199884
// MI455X (gfx1250) — compile-verified
//
#include <hip/hip_runtime.h>
#include <hip/hip_bf16.h>

#define EDGES 98304
#define KK    32
#define NN    3072
#define HH    16
#define SS    32
#define NP1   3073   // N + 1 (graph token row/col)

typedef __attribute__((ext_vector_type(2))) float v2f;
typedef __attribute__((ext_vector_type(8))) float v8f;

// ---------------------------------------------------------------------------
// Kernel 1: stream-fill the bias tensor.
//   out[h,0,j]   = gt[h]
//   out[h,i,0]   = gt[h]            (i >= 1)
//   out[h,i,j]   = logprob[j-1]     (i,j >= 1)  -- edge contributions added later
// 604 MB of stores > 192 MB L2 -> non-temporal stores to avoid thrashing.
// ---------------------------------------------------------------------------
__global__ __launch_bounds__(256) void fill_bias_kernel(
    const float* __restrict__ logp,      // [N]
    const float* __restrict__ gt,        // [H]
    float* __restrict__ out)             // [H, NP1, NP1]
{
    const int j = blockIdx.x * 256 + threadIdx.x;
    if (j >= NP1) return;
    const int i = blockIdx.y;
    const int h = blockIdx.z;
    const float val = (i == 0 || j == 0) ? gt[h] : logp[j - 1];
    __builtin_nontemporal_store(
        val, &out[(size_t)h * NP1 * NP1 + (size_t)i * NP1 + (size_t)j]);
}

// ---------------------------------------------------------------------------
// Kernel 2: weighted spatial embedding via one-hot + WMMA, then scatter-add.
//   acc[e,s] = sum_k w[e,k] * (t[e,k] == s)          (LDS ds_add_f32)
//   edge_feat[16x16] = acc[16x32] @ W_spatial[32x16] (8x V_WMMA_F32_16X16X4_F32)
//   out[h, src+1, dst+1] += edge_feat[e,h]           (global_atomic_add_f32)
// One wave handles 16 edges; 8 waves per block -> 128 edges/block; 768 blocks.
// ---------------------------------------------------------------------------
__global__ __launch_bounds__(256) void edge_wmma_scatter_kernel(
    const int*   __restrict__ types,     // [E, K]
    const float* __restrict__ wts,       // [E, K]
    const int*   __restrict__ gidx,      // [2, E]
    const float* __restrict__ Wsp,       // [S, H] row-major
    float* __restrict__ out)             // [H, NP1, NP1]
{
    __shared__ float acc[8][16][SS];     // per-wave 16 edges x 32 buckets (16 KB)

    const int lane = threadIdx.x & 31;
    const int wave = threadIdx.x >> 5;
    const int eb   = (blockIdx.x * 8 + wave) * 16;   // first edge of this wave

    // zero this wave's accumulator tile
    {
        float* a = &acc[wave][0][0];
        for (int i = lane; i < 16 * SS; i += 32) a[i] = 0.0f;
    }
    __syncthreads();

    // bucket-scatter: lane -> edge (lane&15), half of K (lane>>4)
    {
        const int e  = eb + (lane & 15);
        const int kh = (lane >> 4) * (KK / 2);
        const int*   tp = types + (size_t)e * KK + kh;
        const float* wp = wts   + (size_t)e * KK + kh;
        float* arow = &acc[wave][lane & 15][0];
        #pragma unroll
        for (int k = 0; k < KK / 2; ++k)
            atomicAdd(&arow[tp[k]], wp[k]);       // ds_add_f32 (no return)
    }
    __syncthreads();

    // 8x V_WMMA_F32_16X16X4_F32: D[16x16] = acc[16x32] x Wsp[32x16]
    // A 16x4 f32 layout: lanes {0-15,16-31} hold M=lane&15; VGPR0={K0,K2}, VGPR1={K1,K3}
    // B 4x16  f32 layout: lanes {0-15,16-31} hold N=lane&15; VGPR0={K0,K2}, VGPR1={K1,K3}
    const int mn   = lane & 15;          // M for A / N for B / head column for D
    const int half = lane >> 4;
    v8f c = {};
    #pragma unroll
    for (int k0 = 0; k0 < SS; k0 += 4) {
        const int ka = k0 + half * 2;
        v2f a, b;
        a.x = acc[wave][mn][ka + 0];
        a.y = acc[wave][mn][ka + 1];
        b.x = Wsp[(ka + 0) * HH + mn];
        b.y = Wsp[(ka + 1) * HH + mn];
        c = __builtin_amdgcn_wmma_f32_16x16x4_f32(
                /*neg_a=*/false, a, /*neg_b=*/false, b,
                /*c_mod=*/(short)0, c, /*reuse_a=*/false, /*reuse_b=*/false);
    }

    // D layout: VGPR v holds M = v + 8*half, N = lane&15. Scatter-add to output.
    const int h = mn;
    #pragma unroll
    for (int v = 0; v < 8; ++v) {
        const int e   = eb + v + half * 8;
        const int src = gidx[e];             // graph_index[0, e]
        const int dst = gidx[EDGES + e];     // graph_index[1, e]
        atomicAdd(&out[(size_t)h * NP1 * NP1 +
                       (size_t)(src + 1) * NP1 + (size_t)(dst + 1)], c[v]);
    }
}

// ---------------------------------------------------------------------------
// Inputs (setup_inputs order):
//   0: spatial_types         int32  [E,K]
//   1: spatial_types_weights f32    [E,K]
//   2: graph_index           int32  [2,E]
//   3: batch                 int32  [N]     (unused; single graph)
//   4: node_logprob          f32    [N]
//   5: W_spatial             f32    [S,H]
//   6: graph_token           f32    [1,H,1]
// Output: f32 [H, N+1, N+1]
// ---------------------------------------------------------------------------
extern "C" void kernel_launch(void* const* d_in, const int* in_sizes, int n_in,
                              void* d_out, int out_size, void* d_ws, size_t ws_size,
                              hipStream_t stream) {
    const int*   types = (const int*)  d_in[0];
    const float* wts   = (const float*)d_in[1];
    const int*   gidx  = (const int*)  d_in[2];
    const float* logp  = (const float*)d_in[4];
    const float* Wsp   = (const float*)d_in[5];
    const float* gt    = (const float*)d_in[6];
    float* out = (float*)d_out;

    // Pass 1: fill token row/col + broadcast key log-prob (bandwidth bound).
    dim3 g1((NP1 + 255) / 256, NP1, HH);
    fill_bias_kernel<<<g1, 256, 0, stream>>>(logp, gt, out);

    // Pass 2: WMMA embedding-bag + scatter-add of edge features.
    edge_wmma_scatter_kernel<<<EDGES / 128, 256, 0, stream>>>(types, wts, gidx, Wsp, out);
}